// EnhancedUnitaryRNN_43903155700411
// MI455X (gfx1250) — compile-verified
//
#include <hip/hip_runtime.h>
#include <math.h>

typedef __attribute__((ext_vector_type(2))) float v2f;
typedef __attribute__((ext_vector_type(8))) float v8f;

__device__ __forceinline__ v8f wmma_f32(v2f a, v2f b, v8f c) {
  // 8 args: (neg_a, A, neg_b, B, c_mod, C, reuse_a, reuse_b)
  return __builtin_amdgcn_wmma_f32_16x16x4_f32(false, a, false, b, (short)0, c, false, false);
}

#define TB   256
#define BM   64
#define BN   128
#define BK   16
#define ASTR 17    // padded LDS stride for A (64x16): conflict-free frag reads
#define BSTR 136   // padded LDS stride for B (16x128): lanes 0-15 (K) vs 16-31 (K+2) hit distinct banks

// C[M,N] (+)= alpha * A[M,K] x B[K,N], row-major.
// HA: A has imag part; HB: B has imag part; HC: C has imag part.
// Compile-time specialization keeps the inner loop branch-free (pure ds_load + v_wmma).
template <bool HA, bool HB, bool HC>
__global__ __launch_bounds__(TB) void gemm_kernel(
    const float* Are, const float* Aim,
    const float* Bre, const float* Bim,
    float* Cre, float* Cim,
    int M, int N, int K, float alpha, int accumulate)
{
  __shared__ float sAre[BM * ASTR];
  __shared__ float sAim[HA ? BM * ASTR : 1];
  __shared__ float sBre[BK * BSTR];
  __shared__ float sBim[HB ? BK * BSTR : 1];

  const int tid  = threadIdx.x;
  const int wave = tid >> 5;
  const int lane = tid & 31;
  const int wm   = wave & 1;    // 2 waves along M
  const int wn   = wave >> 1;   // 4 waves along N
  const int lh   = lane >> 4;   // lane half (selects K pair / M+8 group)
  const int lm   = lane & 15;
  const int bm0  = blockIdx.y * BM;
  const int bn0  = blockIdx.x * BN;

  v8f acc_re[2][2], acc_im[2][2];
#pragma unroll
  for (int i = 0; i < 2; i++)
#pragma unroll
    for (int j = 0; j < 2; j++)
#pragma unroll
      for (int e = 0; e < 8; e++) { acc_re[i][j][e] = 0.f; acc_im[i][j][e] = 0.f; }

  for (int kb = 0; kb < K; kb += BK) {
    const bool pf = (kb + BK) < K;
    for (int idx = tid; idx < BM * BK; idx += TB) {
      int r = idx >> 4, c = idx & (BK - 1);
      size_t g = (size_t)(bm0 + r) * K + (kb + c);
      sAre[r * ASTR + c] = Are[g];
      if (HA) sAim[r * ASTR + c] = Aim[g];
      if (pf) {
        __builtin_prefetch(&Are[g + BK], 0, 0);        // next K-panel -> global_prefetch
        if (HA) __builtin_prefetch(&Aim[g + BK], 0, 0);
      }
    }
    for (int idx = tid; idx < BK * BN; idx += TB) {
      int r = idx >> 7, c = idx & (BN - 1);
      size_t g = (size_t)(kb + r) * N + (bn0 + c);
      sBre[r * BSTR + c] = Bre[g];
      if (HB) sBim[r * BSTR + c] = Bim[g];
      if (pf) {
        __builtin_prefetch(&Bre[g + (size_t)BK * N], 0, 0);
        if (HB) __builtin_prefetch(&Bim[g + (size_t)BK * N], 0, 0);
      }
    }
    __syncthreads();

#pragma unroll
    for (int kk = 0; kk < BK; kk += 4) {
      const int kc = kk + lh * 2;  // A/B f32 frag layout: lane half selects K pair {kc,kc+1}
      v2f ar[2], ai[2], br[2], bi[2];
#pragma unroll
      for (int t = 0; t < 2; t++) {
        int m = wm * 32 + t * 16 + lm;
        ar[t].x = sAre[m * ASTR + kc];
        ar[t].y = sAre[m * ASTR + kc + 1];
        if (HA) { ai[t].x = sAim[m * ASTR + kc]; ai[t].y = sAim[m * ASTR + kc + 1]; }
        int n = wn * 32 + t * 16 + lm;
        br[t].x = sBre[kc * BSTR + n];
        br[t].y = sBre[(kc + 1) * BSTR + n];
        if (HB) { bi[t].x = sBim[kc * BSTR + n]; bi[t].y = sBim[(kc + 1) * BSTR + n]; }
      }
#pragma unroll
      for (int tm = 0; tm < 2; tm++)
#pragma unroll
        for (int tn = 0; tn < 2; tn++) {
          acc_re[tm][tn] = wmma_f32(ar[tm], br[tn], acc_re[tm][tn]);
          if (HA && HB)
            acc_re[tm][tn] = wmma_f32(-ai[tm], bi[tn], acc_re[tm][tn]); // re -= im*im
          if (HC) {
            if (HB) acc_im[tm][tn] = wmma_f32(ar[tm], bi[tn], acc_im[tm][tn]);
            if (HA) acc_im[tm][tn] = wmma_f32(ai[tm], br[tn], acc_im[tm][tn]);
          }
        }
    }
    __syncthreads();
  }

#pragma unroll
  for (int tm = 0; tm < 2; tm++)
#pragma unroll
    for (int tn = 0; tn < 2; tn++) {
      int mb = bm0 + wm * 32 + tm * 16 + lh * 8;  // C layout: VGPR e -> row e (+8 for lane half)
      int n  = bn0 + wn * 32 + tn * 16 + lm;
#pragma unroll
      for (int e = 0; e < 8; e++) {
        size_t off = (size_t)(mb + e) * N + n;
        float vr = alpha * acc_re[tm][tn][e];
        Cre[off] = accumulate ? (Cre[off] + vr) : vr;
        if (HC) {
          float vi = alpha * acc_im[tm][tn][e];
          Cim[off] = accumulate ? (Cim[off] + vi) : vi;
        }
      }
    }
}

// A^T where A = i*(M + M^H), pre-scaled by sc; expm(A^T) = expm(A)^T = U^T.
__global__ void build_At_kernel(const float* Mre, const float* Mim,
                                float* Atre, float* Atim, int S, float sc) {
  size_t idx = (size_t)blockIdx.x * TB + threadIdx.x;
  if (idx >= (size_t)S * S) return;
  int i = (int)(idx / S), j = (int)(idx % S);
  Atre[idx] = (Mim[(size_t)i * S + j] - Mim[(size_t)j * S + i]) * sc;
  Atim[idx] = (Mre[(size_t)i * S + j] + Mre[(size_t)j * S + i]) * sc;
}

__global__ void set_identity_kernel(float* Tre, float* Tim, int S) {
  size_t idx = (size_t)blockIdx.x * TB + threadIdx.x;
  if (idx >= (size_t)S * S) return;
  int i = (int)(idx / S), j = (int)(idx % S);
  Tre[idx] = (i == j) ? 1.f : 0.f;
  Tim[idx] = 0.f;
}

// F[k][d] = exp(-2*pi*i*k*d/D)  (symmetric)
__global__ void dft_kernel(float* Fre, float* Fim, int D) {
  size_t idx = (size_t)blockIdx.x * TB + threadIdx.x;
  if (idx >= (size_t)D * D) return;
  int k = (int)(idx / D), d = (int)(idx % D);
  int r = (int)(((long long)k * d) % D);
  float th = -6.283185307179586f * (float)r / (float)D;
  Fre[idx] = cosf(th);
  Fim[idx] = sinf(th);
}

// input_matrix [S,2D] -> W1^T into (W1re,W1im) [D,S], W2^T into (W2re,W2im) [D,S]
__global__ void transpose_halves_kernel(const float* IMre, const float* IMim,
                                        float* W1re, float* W1im,
                                        float* W2re, float* W2im, int S, int D) {
  size_t idx = (size_t)blockIdx.x * TB + threadIdx.x;
  if (idx >= (size_t)D * S) return;
  int d = (int)(idx / S), s = (int)(idx % S);
  size_t row = (size_t)s * 2 * D;
  W1re[idx] = IMre[row + d];
  W1im[idx] = IMim[row + d];
  W2re[idx] = IMre[row + D + d];
  W2im[idx] = IMim[row + D + d];
}

__global__ void modrelu_kernel(float* re, float* im, const float* bias, size_t n) {
  size_t idx = (size_t)blockIdx.x * TB + threadIdx.x;
  if (idx >= n) return;
  float r = re[idx], i = im[idx];
  float mag = sqrtf(r * r + i * i);
  float sc = fmaxf(mag + bias[0], 0.f) / fmaxf(mag, 1e-30f);
  re[idx] = r * sc;
  im[idx] = i * sc;
}

__global__ void modrelu_store_kernel(float* re, float* im, const float* bias,
                                     float* out2, size_t n) {
  size_t idx = (size_t)blockIdx.x * TB + threadIdx.x;
  if (idx >= n) return;
  float r = re[idx], i = im[idx];
  float mag = sqrtf(r * r + i * i);
  float sc = fmaxf(mag + bias[0], 0.f) / fmaxf(mag, 1e-30f);
  r *= sc; i *= sc;
  re[idx] = r; im[idx] = i;
  out2[2 * idx]     = r;   // complex64 memory layout: interleaved re,im
  out2[2 * idx + 1] = i;
}

__global__ void bias_init_kernel(float* out, const float* b, int B, int O) {
  size_t idx = (size_t)blockIdx.x * TB + threadIdx.x;
  if (idx >= (size_t)B * O) return;
  out[idx] = b[idx % O];
}

// Dispatch to the right specialization based on which imag planes exist.
static inline void gemm(hipStream_t st, const float* Are, const float* Aim,
                        const float* Bre, const float* Bim, float* Cre, float* Cim,
                        int M, int N, int K, float alpha, int acc) {
  dim3 g((unsigned)(N / BN), (unsigned)(M / BM));
  if (Aim && Bim && Cim)
    gemm_kernel<true, true, true><<<g, TB, 0, st>>>(Are, Aim, Bre, Bim, Cre, Cim, M, N, K, alpha, acc);
  else if (!Aim && Bim && Cim)
    gemm_kernel<false, true, true><<<g, TB, 0, st>>>(Are, Aim, Bre, Bim, Cre, Cim, M, N, K, alpha, acc);
  else
    gemm_kernel<false, false, false><<<g, TB, 0, st>>>(Are, Aim, Bre, Bim, Cre, Cim, M, N, K, alpha, acc);
}

static inline dim3 ew(size_t n) { return dim3((unsigned)((n + TB - 1) / TB)); }

extern "C" void kernel_launch(void* const* d_in, const int* in_sizes, int n_in,
                              void* d_out, int out_size, void* d_ws, size_t ws_size,
                              hipStream_t stream) {
  (void)n_in; (void)out_size; (void)ws_size;
  const float* X    = (const float*)d_in[0];   // [B,D]
  const float* STre = (const float*)d_in[1];   // [B,S]
  const float* STim = (const float*)d_in[2];
  const float* MSre = (const float*)d_in[3];   // [S,S]
  const float* MSim = (const float*)d_in[4];
  const float* MPre = (const float*)d_in[5];
  const float* MPim = (const float*)d_in[6];
  const float* IMre = (const float*)d_in[7];   // [S,2D]
  const float* IMim = (const float*)d_in[8];
  const float* b1   = (const float*)d_in[9];
  const float* b2   = (const float*)d_in[10];
  const float* Wout = (const float*)d_in[11];  // [2S,O]
  const float* bout = (const float*)d_in[12];  // [O]

  const int S   = (int)(sqrt((double)in_sizes[3]) + 0.5);
  const int Bsz = in_sizes[1] / S;
  const int D   = in_sizes[0] / Bsz;
  const int O   = in_sizes[12];

  const size_t SS = (size_t)S * S;
  const size_t DS = (size_t)D * S;
  const size_t DD = (size_t)D * D;
  const size_t BS = (size_t)Bsz * S;

  float* w = (float*)d_ws;
  float* At_re = w; w += SS;  float* At_im = w; w += SS;
  float* T0re  = w; w += SS;  float* T0im  = w; w += SS;
  float* T1re  = w; w += SS;  float* T1im  = w; w += SS;
  float* USre  = w; w += SS;  float* USim  = w; w += SS;   // U_state^T
  float* UPre  = w; w += SS;  float* UPim  = w; w += SS;   // U_step^T
  float* Fre   = w; w += DD;  float* Fim   = w; w += DD;
  float* W2re  = w; w += DS;  float* W2im  = w; w += DS;
  float* WEre  = w; w += DS;  float* WEim  = w; w += DS;   // W_eff^T = W1^T + F*W2^T
  float* Zre   = w; w += BS;  float* Zim   = w; w += BS;
  float* Yre   = w; w += BS;  float* Yim   = w; w += BS;

  float* out0 = (float*)d_out;            // [B,O] float32
  float* out1 = out0 + (size_t)Bsz * O;   // [B,S] complex64 (interleaved)

  // ---- fold FFT into input projection: W_eff^T = W1^T + F x W2^T (F symmetric) ----
  dft_kernel<<<ew(DD), TB, 0, stream>>>(Fre, Fim, D);
  transpose_halves_kernel<<<ew(DS), TB, 0, stream>>>(IMre, IMim, WEre, WEim, W2re, W2im, S, D);
  gemm(stream, Fre, Fim, W2re, W2im, WEre, WEim, D, S, D, 1.0f, 1);

  // ---- expm(i*H)^T for state & step via scaling(2^-6) + Taylor(K=12) Horner + 6 squarings ----
  const int   KT = 12;
  const int   NSQ = 6;
  const float sc = 1.0f / 64.0f;  // 2^-6
  for (int u = 0; u < 2; ++u) {
    const float* Mre = u ? MPre : MSre;
    const float* Mim = u ? MPim : MSim;
    float* Ure = u ? UPre : USre;
    float* Uim = u ? UPim : USim;
    build_At_kernel<<<ew(SS), TB, 0, stream>>>(Mre, Mim, At_re, At_im, S, sc);
    set_identity_kernel<<<ew(SS), TB, 0, stream>>>(T0re, T0im, S);
    float *are = T0re, *aim = T0im, *bre = T1re, *bim = T1im, *t;
    for (int j = KT; j >= 1; --j) {               // T <- I + (A/j)*T
      set_identity_kernel<<<ew(SS), TB, 0, stream>>>(bre, bim, S);
      gemm(stream, At_re, At_im, are, aim, bre, bim, S, S, S, 1.0f / (float)j, 1);
      t = are; are = bre; bre = t;  t = aim; aim = bim; bim = t;
    }
    for (int q = 0; q < NSQ - 1; ++q) {           // 5 ping-pong squarings
      gemm(stream, are, aim, are, aim, bre, bim, S, S, S, 1.0f, 0);
      t = are; are = bre; bre = t;  t = aim; aim = bim; bim = t;
    }
    gemm(stream, are, aim, are, aim, Ure, Uim, S, S, S, 1.0f, 0);  // final squaring -> U^T
  }

  // ---- z = state x U_state^T + x x W_eff^T ----
  gemm(stream, STre, STim, USre, USim, Zre, Zim, Bsz, S, S, 1.0f, 0);
  gemm(stream, X, nullptr, WEre, WEim, Zre, Zim, Bsz, S, D, 1.0f, 1);
  modrelu_kernel<<<ew(BS), TB, 0, stream>>>(Zre, Zim, b1, BS);

  // ---- next_states = modrelu(z x U_step^T) ----
  gemm(stream, Zre, Zim, UPre, UPim, Yre, Yim, Bsz, S, S, 1.0f, 0);
  modrelu_store_kernel<<<ew(BS), TB, 0, stream>>>(Yre, Yim, b2, out1, BS);

  // ---- output = [re|im] x W_out + b_out ----
  bias_init_kernel<<<ew((size_t)Bsz * O), TB, 0, stream>>>(out0, bout, Bsz, O);
  gemm(stream, Yre, nullptr, Wout, nullptr, out0, nullptr, Bsz, O, S, 1.0f, 1);
  gemm(stream, Yim, nullptr, Wout + (size_t)S * O, nullptr, out0, nullptr, Bsz, O, S, 1.0f, 1);
}